// multiTimeAttention_4440996184171
// MI455X (gfx1250) — compile-verified
//
#include <hip/hip_runtime.h>

#define B_   8
#define LQ_  128
#define LK_  512
#define D_   32
#define ET_  128
#define H_   8
#define ETK_ 16
#define NH_  256

typedef __attribute__((ext_vector_type(2))) float v2f;
typedef __attribute__((ext_vector_type(8))) float v8f;

// D(16x16,f32) = A(16x4,f32) x B(4x16,f32) + C   -- native CDNA5 f32 WMMA
__device__ __forceinline__ v8f wmma4(v2f a, v2f b, v8f c) {
  return __builtin_amdgcn_wmma_f32_16x16x4_f32(
      /*neg_a=*/false, a, /*neg_b=*/false, b,
      /*c_mod=*/(short)0, c, /*reuse_a=*/false, /*reuse_b=*/false);
}

// C[M,N] = A[M,K] @ W[N,K]^T + bias[N]   (torch Linear). One 16x16 tile per wave.
__global__ __launch_bounds__(256) void gemm_nt_bias(
    const float* __restrict__ A, const float* __restrict__ W,
    const float* __restrict__ bias, float* __restrict__ C,
    int M, int N, int K) {
  const int wid  = blockIdx.x * 8 + (threadIdx.x >> 5);
  const int lane = threadIdx.x & 31;
  const int lrow = lane & 15;
  const int half = lane >> 4;
  const int ntiles = N >> 4;
  const int mt = wid / ntiles;
  const int nt = wid % ntiles;
  if (mt * 16 >= M) return;

  const float* arow = A + (size_t)(mt * 16 + lrow) * K;  // A row for this lane
  const float* wrow = W + (size_t)(nt * 16 + lrow) * K;  // W row -> B^T column

  v8f acc = {};
#pragma unroll 8
  for (int k = 0; k < K; k += 4) {
    v2f a, b;
    // A-frag: lane holds A[row][k + 2*half + {0,1}]
    a.x = arow[k + 2 * half];
    a.y = arow[k + 2 * half + 1];
    // B-frag: B[kk][n] = W[n0+n][k+kk]; lane holds kk = 2*half + {0,1}
    b.x = wrow[k + 2 * half];
    b.y = wrow[k + 2 * half + 1];
    acc = wmma4(a, b, acc);
  }
#pragma unroll
  for (int v = 0; v < 8; ++v) {
    const int m = mt * 16 + v + 8 * half;  // C/D layout: vgpr v -> row v (+8 for hi half)
    const int n = nt * 16 + lrow;
    C[(size_t)m * N + n] = acc[v] + bias[n];
  }
}

// One block per (b,h); 8 waves x 16 q-rows. Registers hold full 16x512 score strip.
__global__ __launch_bounds__(256) void attn_kernel(
    const float* __restrict__ qp, const float* __restrict__ kp,
    const float* __restrict__ value, const int* __restrict__ mask,
    float* __restrict__ xm) {
  const int b = blockIdx.x / H_;
  const int h = blockIdx.x % H_;
  const int tid  = threadIdx.x;
  const int wave = tid >> 5;
  const int lane = tid & 31;
  const int lrow = lane & 15;
  const int half = lane >> 4;

  __shared__ float mv[LK_][2 * D_ + 1];   // [k][0:32]=mask*value, [32:64]=mask (padded)
  __shared__ float es[8][16][17];         // per-wave C->A transpose staging (padded)

  // Build masked-value | mask matrix once per block.
  for (int idx = tid; idx < LK_ * D_; idx += 256) {
    const int k = idx >> 5;
    const int d = idx & 31;
    const int mk = mask[((size_t)b * LK_ + k) * D_ + d];
    const float vv = value[((size_t)b * LK_ + k) * D_ + d];
    mv[k][d]       = mk ? vv : 0.0f;
    mv[k][D_ + d]  = (float)mk;
  }
  __syncthreads();

  // ---- Stage 1: S = Qh(16x16) @ Kh^T(16x512), scaled by 1/sqrt(ETK) ----
  v2f aq[4];
  const float* qrow = qp + ((size_t)(b * LQ_ + wave * 16 + lrow)) * ET_ + h * ETK_;
#pragma unroll
  for (int kc = 0; kc < 4; ++kc) {
    aq[kc].x = qrow[kc * 4 + 2 * half];
    aq[kc].y = qrow[kc * 4 + 2 * half + 1];
  }

  v8f sacc[32];
  {
    v8f z = {};
#pragma unroll
    for (int t = 0; t < 32; ++t) sacc[t] = z;
  }
#pragma unroll
  for (int t = 0; t < 32; ++t) {
    const float* krow = kp + ((size_t)(b * LK_ + t * 16 + lrow)) * ET_ + h * ETK_;
#pragma unroll
    for (int kc = 0; kc < 4; ++kc) {
      v2f bk;
      bk.x = krow[kc * 4 + 2 * half];
      bk.y = krow[kc * 4 + 2 * half + 1];
      sacc[t] = wmma4(aq[kc], bk, sacc[t]);
    }
  }

  // ---- Softmax weights: e = exp(0.25*s - rowmax) (masked entries weigh 0) ----
#pragma unroll
  for (int t = 0; t < 32; ++t)
#pragma unroll
    for (int v = 0; v < 8; ++v) sacc[t][v] *= 0.25f;

  float rmax[8];
#pragma unroll
  for (int v = 0; v < 8; ++v) {
    float m = -1e30f;
#pragma unroll
    for (int t = 0; t < 32; ++t) m = fmaxf(m, sacc[t][v]);
    // reduce across the 16 lanes of this half (row m lives in one half)
#pragma unroll
    for (int off = 1; off < 16; off <<= 1) m = fmaxf(m, __shfl_xor(m, off, 32));
    rmax[v] = m;
  }
#pragma unroll
  for (int t = 0; t < 32; ++t)
#pragma unroll
    for (int v = 0; v < 8; ++v) sacc[t][v] = __expf(sacc[t][v] - rmax[v]);

  // ---- Stage 2: [num|den](16x64) = E(16x512) @ mv(512x64) ----
  v8f xacc[4];
  {
    v8f z = {};
#pragma unroll
    for (int c = 0; c < 4; ++c) xacc[c] = z;
  }
#pragma unroll
  for (int t = 0; t < 32; ++t) {
    // spill E tile (C/D layout) to LDS, reload in A layout (in-wave, DS in-order)
#pragma unroll
    for (int v = 0; v < 8; ++v) es[wave][v + 8 * half][lrow] = sacc[t][v];
#pragma unroll
    for (int kk = 0; kk < 4; ++kk) {
      v2f ae;
      ae.x = es[wave][lrow][kk * 4 + 2 * half];
      ae.y = es[wave][lrow][kk * 4 + 2 * half + 1];
      const int kg = t * 16 + kk * 4;
#pragma unroll
      for (int ct = 0; ct < 4; ++ct) {
        v2f bm;
        bm.x = mv[kg + 2 * half][ct * 16 + lrow];
        bm.y = mv[kg + 2 * half + 1][ct * 16 + lrow];
        xacc[ct] = wmma4(ae, bm, xacc[ct]);
      }
    }
  }

  // ---- x = num/den, store merged-head layout xm[b][q][h*D + d] ----
#pragma unroll
  for (int ct = 0; ct < 2; ++ct)
#pragma unroll
    for (int v = 0; v < 8; ++v) {
      const float x = xacc[ct][v] / xacc[ct + 2][v];
      const int q = wave * 16 + v + 8 * half;
      const int d = ct * 16 + lrow;
      xm[((size_t)b * LQ_ + q) * NH_ + h * D_ + d] = x;
    }
}

extern "C" void kernel_launch(void* const* d_in, const int* in_sizes, int n_in,
                              void* d_out, int out_size, void* d_ws, size_t ws_size,
                              hipStream_t stream) {
  const float* query = (const float*)d_in[0];
  const float* key   = (const float*)d_in[1];
  const float* value = (const float*)d_in[2];
  const int*   mask  = (const int*)d_in[3];
  const float* Wq = (const float*)d_in[4];
  const float* bq = (const float*)d_in[5];
  const float* Wk = (const float*)d_in[6];
  const float* bk = (const float*)d_in[7];
  const float* Wo = (const float*)d_in[8];
  const float* bo = (const float*)d_in[9];
  float* out = (float*)d_out;

  float* qp = (float*)d_ws;                        // [B*LQ, ET]  512 KB
  float* kp = qp + (size_t)B_ * LQ_ * ET_;         // [B*LK, ET]  2 MB
  float* xm = kp + (size_t)B_ * LK_ * ET_;         // [B*LQ, NH]  1 MB

  // Q / K projections: tiles = (M/16)*(N/16), 8 waves per block
  gemm_nt_bias<<<(B_ * LQ_ / 16) * (ET_ / 16) / 8, 256, 0, stream>>>(
      query, Wq, bq, qp, B_ * LQ_, ET_, ET_);
  gemm_nt_bias<<<(B_ * LK_ / 16) * (ET_ / 16) / 8, 256, 0, stream>>>(
      key, Wk, bk, kp, B_ * LK_, ET_, ET_);

  attn_kernel<<<B_ * H_, 256, 0, stream>>>(qp, kp, value, mask, xm);

  gemm_nt_bias<<<(B_ * LQ_ / 16) * (NH_ / 16) / 8, 256, 0, stream>>>(
      xm, Wo, bo, out, B_ * LQ_, NH_, NH_);
}